// HashGrid_76527727280771
// MI455X (gfx1250) — compile-verified
//
#include <hip/hip_runtime.h>

// Instant-NGP hash-grid encoding, forward only.
// N=262144 points, L=16 levels, F=2 feats, T=2^19 entries/level (table = 64 MB,
// L2-resident on MI455X's 192 MB L2). Gather-bound: 256 MB of random 8B gathers.
// CDNA5 paths used:
//   - async LDS->global stores (GLOBAL_STORE_ASYNC_FROM_LDS_B128 / ASYNCcnt /
//     s_wait_asynccnt), non-temporal, for coalesced output of per-point 128B rows
//   - saddr + scale_offset gathers: masked hash feeds the load directly, no
//     per-gather VALU address math (level base folded into the SGPR pointer)

namespace {
constexpr int      kLevels = 16;
constexpr unsigned kT      = 1u << 19;
constexpr unsigned kMask   = kT - 1u;
constexpr unsigned kP1     = 2654435761u;
constexpr unsigned kP2     = 805459861u;
constexpr int      kBlock  = 256;
}

__global__ __launch_bounds__(kBlock)
void hashgrid_fwd_kernel(const float* __restrict__ x,
                         const float* __restrict__ table,
                         float* __restrict__ out,
                         int n)
{
    // torch-ngp / tcnn growth: ceil(16 * (128^(1/15))^l)
    constexpr float RESF[kLevels] = {
        16.f, 23.f, 31.f, 43.f, 59.f, 81.f, 112.f, 154.f,
        213.f, 295.f, 407.f, 562.f, 777.f, 1073.f, 1483.f, 2048.f};

    // Staging tile: identical layout to the block's slice of `out` ->
    // async stores are a pure linear LDS->global copy, fully coalesced.
    __shared__ float2 lds_out[kBlock * 16];   // 32 KB of the 320 KB/WGP

    const int tid = threadIdx.x;
    const int p   = blockIdx.x * kBlock + tid;

    // Invalid lanes compute on (0,0,0): defined, in-range, keeps EXEC full.
    float cx = 0.f, cy = 0.f, cz = 0.f;
    if (p < n) {
        const float* xp = x + (size_t)p * 3;   // 12B/lane -> global_load_b96
        cx = xp[0]; cy = xp[1]; cz = xp[2];
    }

#pragma unroll
    for (int l = 0; l < kLevels; ++l) {
        // Level base folded into the pointer: uniform, SALU-only, lets each
        // gather be  global_load_b64 v, v_idx, s[base] scale_offset.
        const float2* __restrict__ tbl =
            (const float2*)table + (size_t)((unsigned)l * kT);

        const float r = RESF[l];
        const float px = cx * r, py = cy * r, pz = cz * r;
        const float fx = floorf(px), fy = floorf(py), fz = floorf(pz);
        const float wx = px - fx, wy = py - fy, wz = pz - fz;

        const unsigned ix = (unsigned)(int)fx;
        const unsigned iy = (unsigned)(int)fy;
        const unsigned iz = (unsigned)(int)fz;

        // spatial hash, products factored (prime for x is 1)
        const unsigned hx0 = ix,       hx1 = ix + 1u;
        const unsigned hy0 = iy * kP1, hy1 = hy0 + kP1;
        const unsigned hz0 = iz * kP2, hz1 = hz0 + kP2;

        // 8 independent 8-byte gathers (global_load_b64), L2-resident table
        const float2 c000 = tbl[(hx0 ^ hy0 ^ hz0) & kMask];
        const float2 c001 = tbl[(hx0 ^ hy0 ^ hz1) & kMask];
        const float2 c010 = tbl[(hx0 ^ hy1 ^ hz0) & kMask];
        const float2 c011 = tbl[(hx0 ^ hy1 ^ hz1) & kMask];
        const float2 c100 = tbl[(hx1 ^ hy0 ^ hz0) & kMask];
        const float2 c101 = tbl[(hx1 ^ hy0 ^ hz1) & kMask];
        const float2 c110 = tbl[(hx1 ^ hy1 ^ hz0) & kMask];
        const float2 c111 = tbl[(hx1 ^ hy1 ^ hz1) & kMask];

        // exact trilinear via lerp tree (z, then y, then x); FMA form
        const float z00x = fmaf(wz, c001.x - c000.x, c000.x);
        const float z01x = fmaf(wz, c011.x - c010.x, c010.x);
        const float z10x = fmaf(wz, c101.x - c100.x, c100.x);
        const float z11x = fmaf(wz, c111.x - c110.x, c110.x);
        const float y0x  = fmaf(wy, z01x - z00x, z00x);
        const float y1x  = fmaf(wy, z11x - z10x, z10x);
        const float outx = fmaf(wx, y1x - y0x, y0x);

        const float z00y = fmaf(wz, c001.y - c000.y, c000.y);
        const float z01y = fmaf(wz, c011.y - c010.y, c010.y);
        const float z10y = fmaf(wz, c101.y - c100.y, c100.y);
        const float z11y = fmaf(wz, c111.y - c110.y, c110.y);
        const float y0y  = fmaf(wy, z01y - z00y, z00y);
        const float y1y  = fmaf(wy, z11y - z10y, z10y);
        const float outy = fmaf(wx, y1y - y0y, y0y);

        // Write this level's 8B straight to the LDS tile (ds_store_b64):
        // frees the 32-reg accumulator -> more waves in flight for the gathers.
        lds_out[tid * kLevels + l] = make_float2(outx, outy);
    }
    __syncthreads();

    // Coalesced async drain: 2048 x 16B chunks, lane i of pass k takes chunk
    // tid + k*256 -> consecutive lanes write consecutive 16B of global memory.
    // Non-temporal: keep the 64MB hash table L2-resident, stream the output out.
    float* out_base = out + (size_t)blockIdx.x * kBlock * 32;
    const unsigned lds_base = (unsigned)(size_t)lds_out;  // low 32b = LDS offset
#pragma unroll
    for (int k = 0; k < 8; ++k) {
        const unsigned g = (unsigned)tid + (unsigned)k * kBlock;   // chunk id
        const int point_of_chunk = blockIdx.x * kBlock + (int)(g >> 3);
        if (point_of_chunk < n) {
            const unsigned long long gaddr =
                (unsigned long long)(size_t)(out_base + (size_t)g * 4);
            const unsigned laddr = lds_base + g * 16u;
            asm volatile("global_store_async_from_lds_b128 %0, %1, off th:TH_STORE_NT"
                         :: "v"(gaddr), "v"(laddr)
                         : "memory");
        }
    }
    // Drain ASYNCcnt before wave retires (endpgm also waits, but be explicit).
    asm volatile("s_wait_asynccnt 0" ::: "memory");
}

extern "C" void kernel_launch(void* const* d_in, const int* in_sizes, int n_in,
                              void* d_out, int out_size, void* d_ws, size_t ws_size,
                              hipStream_t stream)
{
    const float* x     = (const float*)d_in[0];   // [N,3] f32
    const float* table = (const float*)d_in[1];   // [L*T, 2] f32
    float*       out   = (float*)d_out;           // [N, 32] f32

    const int n      = in_sizes[0] / 3;
    const int blocks = (n + kBlock - 1) / kBlock;

    hipLaunchKernelGGL(hashgrid_fwd_kernel, dim3(blocks), dim3(kBlock), 0, stream,
                       x, table, out, n);
}